// LSTMCell_28003186770569
// MI455X (gfx1250) — compile-verified
//
#include <hip/hip_runtime.h>
#include <hip/hip_bf16.h>
#include <math.h>

#define SEQ 4096
#define INP 1024
#define HID 1024
#define NG  4
#define G4  (NG * HID)          // 4096 gate outputs per step

typedef float v2f __attribute__((ext_vector_type(2)));
typedef float v8f __attribute__((ext_vector_type(8)));

// ---------------------------------------------------------------------------
// Phase 0: zero h0 and the grid-sync counter (ws is poisoned with 0xAA).
// ---------------------------------------------------------------------------
__global__ void lstm_init(float* __restrict__ hbuf, unsigned* __restrict__ syncc) {
    int i = blockIdx.x * blockDim.x + threadIdx.x;
    if (i < HID) hbuf[i] = 0.0f;
    if (i == 0) *syncc = 0u;
}

// ---------------------------------------------------------------------------
// Phase 1: Gx[t][g*HID + j] = x_t . Wg[0:1024, j] + bias_g[j]
// Dense GEMM on the WMMA pipe, fp32 (V_WMMA_F32_16X16X4_F32).
//
// Block: 256 threads = 8 waves. blockIdx.x fixes (gate, n0); the full B panel
// W[0:1024, n0:n0+16] (64 KB) is staged once into LDS and shared by all 8
// waves (8x global-traffic reduction). Each wave then computes one 16x16
// output tile at its own m0, K-loop in steps of 8 with two independent
// accumulator chains to pipeline WMMA latency. Bias folded into acc init.
//
// A (16x4 f32) per-lane layout:  lanes 0-15: vgpr0=K0,vgpr1=K1 (row M=lane)
//                                lanes16-31: vgpr0=K2,vgpr1=K3 (row M=lane-16)
// B (4x16 f32):                  lanes 0-15: vgpr0=B[K0][N],vgpr1=B[K1][N]
//                                lanes16-31: vgpr0=B[K2][N],vgpr1=B[K3][N]
// C/D (16x16 f32): vgpr r: lanes0-15 -> (M=r, N=lane); lanes16-31 -> (M=8+r)
// ---------------------------------------------------------------------------
__global__ void __launch_bounds__(256)
lstm_gemm_x(const float* __restrict__ X,
            const float* __restrict__ Wf, const float* __restrict__ bf,
            const float* __restrict__ Wi, const float* __restrict__ bi,
            const float* __restrict__ Wm, const float* __restrict__ bm,
            const float* __restrict__ Wo, const float* __restrict__ bo,
            float* __restrict__ Gx) {
    __shared__ float bl[INP * 16];        // B panel: bl[k*16 + n], 64 KB

    const int lane = threadIdx.x & 31;
    const int wave = threadIdx.x >> 5;
    const int l    = lane & 15;           // column/row within tile
    const int half = lane >> 4;           // K half selector

    const int ntile = blockIdx.x;         // 0..255 : gate * 64 + n-tile
    const int gate  = ntile >> 6;
    const int n0    = (ntile & 63) * 16;
    const int m0    = (blockIdx.y * 8 + wave) * 16;   // 8 waves -> 8 M tiles

    const float* W = (gate == 0) ? Wf : (gate == 1) ? Wi : (gate == 2) ? Wm : Wo;
    const float* B = (gate == 0) ? bf : (gate == 1) ? bi : (gate == 2) ? bm : bo;

    // Cooperative stage of the 1024x16 B panel into LDS (64B rows, coalesced).
    for (int idx = threadIdx.x; idx < INP * 16; idx += 256) {
        const int k = idx >> 4;
        const int n = idx & 15;
        bl[idx] = W[(size_t)k * HID + n0 + n];
    }
    __syncthreads();

    const float bv = B[n0 + l];
    v8f acc0 = {bv, bv, bv, bv, bv, bv, bv, bv};
    v8f acc1 = {0.f, 0.f, 0.f, 0.f, 0.f, 0.f, 0.f, 0.f};

    const float* arow = X + (size_t)(m0 + l) * INP;   // A row for this lane

    #pragma unroll 2
    for (int k0 = 0; k0 < INP; k0 += 8) {
        const int ka = k0 + 2 * half;
        v2f a0, b0, a1, b1;
        a0.x = arow[ka];
        a0.y = arow[ka + 1];
        b0.x = bl[ka * 16 + l];
        b0.y = bl[(ka + 1) * 16 + l];
        a1.x = arow[ka + 4];
        a1.y = arow[ka + 5];
        b1.x = bl[(ka + 4) * 16 + l];
        b1.y = bl[(ka + 5) * 16 + l];
        acc0 = __builtin_amdgcn_wmma_f32_16x16x4_f32(
            false, a0, false, b0, (short)0, acc0, false, false);
        acc1 = __builtin_amdgcn_wmma_f32_16x16x4_f32(
            false, a1, false, b1, (short)0, acc1, false, false);
    }
    acc0 = acc0 + acc1;

    const int row_base = m0 + half * 8;
    float* outp = Gx + (size_t)gate * HID + (n0 + l);
    #pragma unroll
    for (int r = 0; r < 8; ++r) {
        outp[(size_t)(row_base + r) * G4] = acc0[r];
    }
}

// ---------------------------------------------------------------------------
// Phase 2: persistent recurrence. 16 blocks x 256 threads.
// Block b owns columns j in [b*64, b*64+64); thread = gate*64 + jj.
// One device-wide sync per timestep via monotonic atomic counter.
// ---------------------------------------------------------------------------
#define RBLK 16

__device__ __forceinline__ float sigmoid_f(float x) {
    return 1.0f / (1.0f + __expf(-x));
}

__device__ __forceinline__ void grid_sync(unsigned* cnt, unsigned target) {
    __syncthreads();
    if (threadIdx.x == 0) {
        __threadfence();
        __atomic_fetch_add(cnt, 1u, __ATOMIC_ACQ_REL);
        while (__atomic_load_n(cnt, __ATOMIC_ACQUIRE) < target) {
            __builtin_amdgcn_s_sleep(2);
        }
    }
    __syncthreads();
    __threadfence();
}

__global__ void __launch_bounds__(256)
lstm_recur(const float* __restrict__ Gx,
           const float* __restrict__ Wf, const float* __restrict__ Wi,
           const float* __restrict__ Wm, const float* __restrict__ Wo,
           float* __restrict__ hbuf, unsigned* __restrict__ syncc,
           float* __restrict__ out) {
    __shared__ float h_lds[HID];
    __shared__ float pre[NG][64];

    const int tid = threadIdx.x;
    const int g   = tid >> 6;        // gate 0..3
    const int jj  = tid & 63;        // column within slice
    const int j0  = blockIdx.x * 64;
    const int j   = j0 + jj;

    const float* Wg = (g == 0) ? Wf : (g == 1) ? Wi : (g == 2) ? Wm : Wo;
    // h-part of the weight: rows INP..INP+HID-1, column j
    const float* Wcol = Wg + (size_t)INP * HID + j;

    float cstate = 0.0f;             // c for column j0+tid (valid for tid<64)

    for (int t = 0; t < SEQ; ++t) {
        // Bring full h_{t-1} into LDS (written by all blocks last step).
        #pragma unroll
        for (int k = tid; k < HID; k += 256) h_lds[k] = hbuf[k];
        __syncthreads();

        // 1024-length dot: coalesced column reads, L2-resident weights.
        float acc = 0.0f;
        #pragma unroll 8
        for (int k = 0; k < HID; ++k) {
            acc = fmaf(h_lds[k], Wcol[(size_t)k * HID], acc);
        }
        pre[g][jj] = acc + Gx[(size_t)t * G4 + (size_t)g * HID + j];
        __syncthreads();

        if (tid < 64) {
            const float fg = sigmoid_f(pre[0][tid]);
            const float ig = sigmoid_f(pre[1][tid]);
            const float cc = tanhf(pre[2][tid]);
            const float og = sigmoid_f(pre[3][tid]);
            cstate = fg * cstate + ig * cc;
            const float h = og * tanhf(cstate);
            hbuf[j0 + tid] = h;
            out[(size_t)t * HID + j0 + tid] = h;
        }
        // Device-wide release/acquire barrier before anyone reads new h.
        grid_sync(syncc, (unsigned)RBLK * (unsigned)(t + 1));
    }
}

// ---------------------------------------------------------------------------
extern "C" void kernel_launch(void* const* d_in, const int* in_sizes, int n_in,
                              void* d_out, int out_size, void* d_ws, size_t ws_size,
                              hipStream_t stream) {
    const float* X  = (const float*)d_in[0];
    const float* Wf = (const float*)d_in[1];
    const float* bf = (const float*)d_in[2];
    const float* Wi = (const float*)d_in[3];
    const float* bi = (const float*)d_in[4];
    const float* Wm = (const float*)d_in[5];
    const float* bm = (const float*)d_in[6];
    const float* Wo = (const float*)d_in[7];
    const float* bo = (const float*)d_in[8];
    float* out = (float*)d_out;

    // Workspace layout: Gx [SEQ][4*HID] f32 (64 MB) | h [HID] f32 | counter
    float*    Gx    = (float*)d_ws;
    float*    hbuf  = Gx + (size_t)SEQ * G4;
    unsigned* syncc = (unsigned*)(hbuf + HID);

    lstm_init<<<4, 256, 0, stream>>>(hbuf, syncc);

    dim3 gridA(NG * HID / 16, SEQ / 16 / 8);   // 256 x 32 blocks, 8 waves each
    lstm_gemm_x<<<gridA, 256, 0, stream>>>(X, Wf, bf, Wi, bi, Wm, bm, Wo, bo, Gx);

    lstm_recur<<<RBLK, 256, 0, stream>>>(Gx, Wf, Wi, Wm, Wo, hbuf, syncc, out);
}